// QNNWrapper_65481071400333
// MI455X (gfx1250) — compile-verified
//
#include <hip/hip_runtime.h>
#include <math.h>

// QNN 20-qubit statevector sim for MI455X (gfx1250, wave32).
// RX/RY layer fused into five 16x16 complex unitaries applied with
// v_wmma_f32_16x16x4_f32; CNOT ring + RZ entangler collapses to a fixed
// GF(2) index permutation (RZ phases cancel in |amp|^2, so weights do not
// affect <Z0>); deterministic tree reduction for the expectation value.

#define NQ   20
#define REPS 3
#define N_STATES (1u << NQ)   // 1,048,576 amplitudes

typedef __attribute__((ext_vector_type(2))) float v2f;
typedef __attribute__((ext_vector_type(8))) float v8f;

// float-offset layout inside d_ws
#define U_RE_OFF 0            // 5*256 floats
#define U_IM_OFF 1280         // 5*256 floats
#define PART_OFF 4096         // 512 partial sums
#define PSI_OFF  8192         // 4 planes of 2^20 floats (Are, Aim, Bre, Bim)
#define PLANE    N_STATES
#define NPART    512

// ---------------------------------------------------------------------------
// Reverse-propagate an output basis index through the 60 CNOTs.
// Forward op k maps psi_after[b] = psi_before[C_k(b)], C_{c,t}: bit_t ^= bit_c.
// psi_final[b] = psi_0[C_1(C_2(...C_60(b)))] -> apply last CNOT first.
// Wire w lives at bit (NQ-1-w) of the flattened index.
__device__ __forceinline__ unsigned qnn_backprop(unsigned x) {
#pragma unroll
  for (int r = REPS - 1; r >= 0; --r) {
#pragma unroll
    for (int i = NQ - 1; i >= 0; --i) {
      const int pc = NQ - 1 - i;
      const int pt = NQ - 1 - ((i + 1) % NQ);
      x ^= ((x >> pc) & 1u) << pt;
    }
  }
  return x;
}

// ---------------------------------------------------------------------------
// Build the five fused 16x16 unitaries U_g = kron of (RY*RX) over wires 4g..4g+3.
// u = RY(h)*RX(h) = [[c^2 + i s^2,  -cs(1+i)],
//                    [cs(1-i),       c^2 - i s^2]],  c=cos(h), s=sin(h), h=x/2.
__global__ __launch_bounds__(256) void qnn_prep(const float* __restrict__ inputs,
                                                float* __restrict__ wsf) {
  const int m = threadIdx.x >> 4;
  const int k = threadIdx.x & 15;
  for (int g = 0; g < 5; ++g) {
    float re = 1.f, im = 0.f;
#pragma unroll
    for (int j = 0; j < 4; ++j) {
      const float h = 0.5f * inputs[4 * g + j];
      const float c = cosf(h), s = sinf(h);
      const int mb = (m >> (3 - j)) & 1;
      const int kb = (k >> (3 - j)) & 1;
      float ur, ui;
      if (mb == 0) {
        if (kb == 0) { ur = c * c;  ui = s * s;  }
        else         { ur = -c * s; ui = -c * s; }
      } else {
        if (kb == 0) { ur = c * s;  ui = -c * s; }
        else         { ur = c * c;  ui = -s * s; }
      }
      const float nr = re * ur - im * ui;
      const float ni = re * ui + im * ur;
      re = nr; im = ni;
    }
    wsf[U_RE_OFF + g * 256 + m * 16 + k] = re;
    wsf[U_IM_OFF + g * 256 + m * 16 + k] = im;
  }
}

// psi = |0...0>
__global__ __launch_bounds__(256) void qnn_init(float* __restrict__ re,
                                                float* __restrict__ im) {
  const unsigned i = blockIdx.x * 256u + threadIdx.x;
  re[i] = (i == 0u) ? 1.f : 0.f;
  im[i] = 0.f;
}

// ---------------------------------------------------------------------------
// Apply 16x16 complex U_g to qubit group g of the statevector.
// Index = (hi << (lw+4)) | (chunk << lw) | low, lw = 16-4g.
// out[m,rest] = sum_k U[m,k] * in[k,rest], as 4 real 16x16x16 matmuls done with
// chained v_wmma_f32_16x16x4_f32 (4 K-steps of K=4 each).
__global__ __launch_bounds__(256) void qnn_apply16(
    const float* __restrict__ wsf, int g,
    const float* __restrict__ srcRe, const float* __restrict__ srcIm,
    float* __restrict__ dstRe, float* __restrict__ dstIm) {
  const int lane = threadIdx.x & 31;
  const int wave = threadIdx.x >> 5;
  const int m    = lane & 15;
  const int koff = (lane >> 4) << 1;   // lanes 16-31 hold K+2 (ISA 16x4 A layout)

  const float* Ur = wsf + U_RE_OFF + g * 256;
  const float* Ui = wsf + U_IM_OFF + g * 256;

  // Hoisted A fragments: Ur, Ui, and -Ui (f32 WMMA has no A-negate modifier).
  v2f aRe[4], aIm[4], aImN[4];
#pragma unroll
  for (int s = 0; s < 4; ++s) {
    const int kb = 4 * s + koff;
    aRe[s][0] = Ur[m * 16 + kb];  aRe[s][1] = Ur[m * 16 + kb + 1];
    aIm[s][0] = Ui[m * 16 + kb];  aIm[s][1] = Ui[m * 16 + kb + 1];
    aImN[s][0] = -aIm[s][0];      aImN[s][1] = -aIm[s][1];
  }

  const int lw = 16 - 4 * g;
  const unsigned lmask = (lw == 0) ? 0u : ((1u << lw) - 1u);
  const int waveGlobal = blockIdx.x * 8 + wave;   // 2048 waves, 4096 tiles

#pragma unroll
  for (int tt = 0; tt < 2; ++tt) {
    const unsigned t    = (unsigned)waveGlobal * 2u + (unsigned)tt;
    const unsigned rest = t * 16u + (unsigned)(lane & 15);   // column n = lane&15
    const unsigned low  = rest & lmask;
    const unsigned hi   = rest >> lw;
    const unsigned base = (hi << (lw + 4)) | low;            // chunk ORed below

    v8f cRe = {}; v8f cIm = {};
#pragma unroll
    for (int s = 0; s < 4; ++s) {
      const unsigned kb = 4u * s + (unsigned)koff;
      const unsigned i0 = base | (kb << lw);
      const unsigned i1 = base | ((kb + 1u) << lw);
      v2f bRe, bIm;
      bRe[0] = srcRe[i0]; bRe[1] = srcRe[i1];
      bIm[0] = srcIm[i0]; bIm[1] = srcIm[i1];
      // Re += Ur*Br - Ui*Bi ; Im += Ur*Bi + Ui*Br
      cRe = __builtin_amdgcn_wmma_f32_16x16x4_f32(false, aRe[s],  false, bRe,
                                                  (short)0, cRe, false, false);
      cRe = __builtin_amdgcn_wmma_f32_16x16x4_f32(false, aImN[s], false, bIm,
                                                  (short)0, cRe, false, false);
      cIm = __builtin_amdgcn_wmma_f32_16x16x4_f32(false, aRe[s],  false, bIm,
                                                  (short)0, cIm, false, false);
      cIm = __builtin_amdgcn_wmma_f32_16x16x4_f32(false, aIm[s],  false, bRe,
                                                  (short)0, cIm, false, false);
    }
    // C/D layout: VGPR v -> M=v (lanes 0-15) / M=v+8 (lanes 16-31), N=lane&15.
    const unsigned mBase = (unsigned)(lane >> 4) * 8u;
#pragma unroll
    for (int v = 0; v < 8; ++v) {
      const unsigned oi = base | ((mBase + (unsigned)v) << lw);
      dstRe[oi] = cRe[v];
      dstIm[oi] = cIm[v];
    }
  }
}

// ---------------------------------------------------------------------------
// Entangler as permutation gather + <Z0> partial reduction (deterministic).
__global__ __launch_bounds__(256) void qnn_entangle_reduce(
    const float* __restrict__ psiRe, const float* __restrict__ psiIm,
    float* __restrict__ partials) {
  __shared__ float red[256];
  float acc = 0.f;
  const unsigned base = blockIdx.x * 2048u + threadIdx.x;
#pragma unroll
  for (int j = 0; j < 8; ++j) {
    const unsigned b = base + (unsigned)j * 256u;
    const unsigned x = qnn_backprop(b);
    const float ar = psiRe[x], ai = psiIm[x];
    const float p = ar * ar + ai * ai;
    acc += ((b >> (NQ - 1)) & 1u) ? -p : p;   // wire 0 = MSB
  }
  red[threadIdx.x] = acc;
  __syncthreads();
  for (int s = 128; s > 0; s >>= 1) {
    if ((int)threadIdx.x < s) red[threadIdx.x] += red[threadIdx.x + s];
    __syncthreads();
  }
  if (threadIdx.x == 0) partials[blockIdx.x] = red[0];
}

__global__ __launch_bounds__(512) void qnn_final(const float* __restrict__ partials,
                                                 float* __restrict__ out) {
  __shared__ float red[512];
  red[threadIdx.x] = partials[threadIdx.x];
  __syncthreads();
  for (int s = 256; s > 0; s >>= 1) {
    if ((int)threadIdx.x < s) red[threadIdx.x] += red[threadIdx.x + s];
    __syncthreads();
  }
  if (threadIdx.x == 0) out[0] = red[0];
}

// ---------------------------------------------------------------------------
// Zero-scratch fallback (used only if d_ws is too small): same math via
// per-group probability tables; P(bit)=c^4+s^4 (bit 0) or 2c^2s^2 (bit 1).
__global__ __launch_bounds__(256) void qnn_fallback(const float* __restrict__ inputs,
                                                    float* __restrict__ out) {
  __shared__ float P[5][16];
  __shared__ float red[256];
  if (threadIdx.x < 80) {
    const int g = threadIdx.x >> 4, v = threadIdx.x & 15;
    float p = 1.f;
#pragma unroll
    for (int j = 0; j < 4; ++j) {
      const float h = 0.5f * inputs[4 * g + j];
      const float c = cosf(h), s = sinf(h);
      const float c2 = c * c, s2 = s * s;
      p *= (((v >> (3 - j)) & 1) != 0) ? 2.f * c2 * s2 : (c2 * c2 + s2 * s2);
    }
    P[g][v] = p;
  }
  __syncthreads();
  float acc = 0.f;
  for (unsigned b = threadIdx.x; b < N_STATES; b += 256u) {
    const unsigned x = qnn_backprop(b);
    const float p = P[0][(x >> 16) & 15] * P[1][(x >> 12) & 15] *
                    P[2][(x >> 8) & 15] * P[3][(x >> 4) & 15] * P[4][x & 15];
    acc += ((b >> (NQ - 1)) & 1u) ? -p : p;
  }
  red[threadIdx.x] = acc;
  __syncthreads();
  for (int s = 128; s > 0; s >>= 1) {
    if ((int)threadIdx.x < s) red[threadIdx.x] += red[threadIdx.x + s];
    __syncthreads();
  }
  if (threadIdx.x == 0) out[0] = red[0];
}

// ---------------------------------------------------------------------------
extern "C" void kernel_launch(void* const* d_in, const int* in_sizes, int n_in,
                              void* d_out, int out_size, void* d_ws, size_t ws_size,
                              hipStream_t stream) {
  (void)in_sizes; (void)n_in; (void)out_size;
  const float* inputs = (const float*)d_in[0];
  // d_in[1] (weights) is provably irrelevant to <Z0>: RZ phases cancel in |amp|^2.
  float* out = (float*)d_out;
  float* wsf = (float*)d_ws;

  const size_t needBytes = (size_t)(PSI_OFF + 4u * PLANE) * sizeof(float);
  if (ws_size >= needBytes) {
    float* Are  = wsf + PSI_OFF;
    float* Aim  = Are + PLANE;
    float* Bre  = Aim + PLANE;
    float* Bim  = Bre + PLANE;
    float* part = wsf + PART_OFF;

    qnn_prep<<<1, 256, 0, stream>>>(inputs, wsf);
    qnn_init<<<N_STATES / 256, 256, 0, stream>>>(Are, Aim);

    const float* sr = Are; const float* si = Aim;
    float* dr = Bre; float* di = Bim;
    for (int g = 0; g < 5; ++g) {
      qnn_apply16<<<256, 256, 0, stream>>>(wsf, g, sr, si, dr, di);
      const float* tr = sr; const float* ti = si;
      sr = dr; si = di;
      dr = (float*)tr; di = (float*)ti;
    }
    // After 5 passes the final state is in (sr, si).
    qnn_entangle_reduce<<<NPART, 256, 0, stream>>>(sr, si, part);
    qnn_final<<<1, 512, 0, stream>>>(part, out);
  } else {
    qnn_fallback<<<1, 256, 0, stream>>>(inputs, out);
  }
}